// PPF_11957188952710
// MI455X (gfx1250) — compile-verified
//
#include <hip/hip_runtime.h>
#include <stdint.h>

typedef __attribute__((ext_vector_type(2))) float v2f;
typedef __attribute__((ext_vector_type(8))) float v8f;

#define NB       4
#define NPTS     4096
#define KNN      32
#define R2       0.01f          /* RADIUS^2 */
#define PPF_EPS  1e-12f
#define NTILES   (NPTS / 16)

/* s_wait_asynccnt via asm: guarantees ordering vs the LDS reads through the
   "memory" clobber (async loads retire in order, so <=1 means the older
   double-buffer stage has landed). */
#define ASYNC_WAIT(n) asm volatile("s_wait_asynccnt " #n ::: "memory")

// One wave32 per (batch b, tile of 16 query rows).
// Distance tiles via V_WMMA_F32_16X16X4_F32 (Gram with K=4, 4th comp zero):
//   d2(M,N) = |q_M|^2 + |c_N|^2 - 2 * (q_M . c_N)
// A layout (32-bit 16x4): lanes 0-15: V0=K0(x), V1=K1(y); lanes 16-31: V0=K2(z), V1=K3(0)
// B layout (32-bit 4x16): mirrored.
// C/D layout: VGPR v -> row v (lanes 0-15) / row v+8 (lanes 16-31), col = lane%16.
// Column coords are double-buffered into LDS with global_load_async_to_lds_b128.
__global__ __launch_bounds__(32) void ppf_ball_wmma(
    const float* __restrict__ coords,
    const float* __restrict__ normals,
    float* __restrict__ out)
{
  __shared__ float colbuf[2][3][16];    // 384 B: double-buffered column slab

  const int  lane = threadIdx.x;
  const int  hl   = lane & 15;          // column / duplicated-row slot
  const bool hi   = lane >= 16;

  const int wg   = blockIdx.x;          // 0 .. NB*NTILES-1
  const int b    = wg / NTILES;
  const int tile = wg % NTILES;

  const float* cb  = coords  + (size_t)b * 3 * NPTS;
  const float* nbp = normals + (size_t)b * 3 * NPTS;

  // Query-row data: lane l holds row (l & 15), duplicated across halves.
  const int   qn  = tile * 16 + hl;
  const float qx  = cb[qn],  qy = cb[NPTS + qn],  qz = cb[2 * NPTS + qn];
  const float nx  = nbp[qn], ny = nbp[NPTS + qn], nz = nbp[2 * NPTS + qn];
  const float sqq = qx * qx + qy * qy + qz * qz;

  // A matrix operand (loop invariant).
  v2f amat;
  amat.x = hi ? qz : qx;
  amat.y = hi ? 0.0f : qy;

  // Tile-invariant per-v broadcasts of "my half's row for C-vgpr v" data.
  float sqqr[8], qxr[8], qyr[8], qzr[8], nxr[8], nyr[8], nzr[8];
#pragma unroll
  for (int v = 0; v < 8; ++v) {
    const int r = hi ? (v + 8) : v;     // row handled by my half in C-vgpr v
    sqqr[v] = __shfl(sqq, r);
    qxr[v]  = __shfl(qx, r);  qyr[v] = __shfl(qy, r);  qzr[v] = __shfl(qz, r);
    nxr[v]  = __shfl(nx, r);  nyr[v] = __shfl(ny, r);  nzr[v] = __shfl(nz, r);
  }

  // Async stage: 12 lanes each copy 16 B (3 planes x 4 chunks) => one
  // global_load_async_to_lds_b128 per tile, tracked by ASYNCcnt.
  const unsigned     ldsbase = (unsigned)(uintptr_t)(void*)&colbuf[0][0][0];
  const int          sp = lane >> 2;            // plane 0..2 (lanes < 12)
  const int          sc = lane & 3;             // 4-float chunk
  const float*       gplane = cb + sp * NPTS + sc * 4;

  auto stage = [&](int t) {
    if (lane < 12) {
      const unsigned l = ldsbase + (unsigned)(((t & 1) * 3 + sp) * 64 + sc * 16);
      const unsigned long long ga = (unsigned long long)(uintptr_t)(gplane + t * 16);
      asm volatile("global_load_async_to_lds_b128 %0, %1, off"
                   :: "v"(l), "v"(ga) : "memory");
    }
  };

  stage(0);

  int   cnt = 0;                        // lane r (<16) owns count for row r
  float mx  = -3.402823466e38f;         // lane r (<16) owns running max angle

  for (int t = 0; t < NTILES; ++t) {
    if (t + 1 < NTILES) {
      stage(t + 1);                     // keep one stage in flight
      ASYNC_WAIT(0x1);                  // older stage (tile t) has landed
    } else {
      ASYNC_WAIT(0x0);
    }

    if (t + 8 < NTILES) {               // warm WGP$/L2 ahead of the async DMA
      __builtin_prefetch(&cb[t * 16 + 128], 0, 3);
      __builtin_prefetch(&cb[NPTS + t * 16 + 128], 0, 3);
      __builtin_prefetch(&cb[2 * NPTS + t * 16 + 128], 0, 3);
    }

    const float cx  = colbuf[t & 1][0][hl];
    const float cy  = colbuf[t & 1][1][hl];
    const float cz  = colbuf[t & 1][2][hl];
    const float sqc = cx * cx + cy * cy + cz * cz;

    v2f bmat;
    bmat.x = hi ? cz : cx;
    bmat.y = hi ? 0.0f : cy;

    v8f acc = {};
    acc = __builtin_amdgcn_wmma_f32_16x16x4_f32(
        /*neg_a=*/false, amat, /*neg_b=*/false, bmat,
        /*c_mod=*/(short)0, acc, /*reuse_a=*/false, /*reuse_b=*/false);

    // Squared distances + cheap tile-level reject via fmin tree (one ballot).
    float d2[8];
#pragma unroll
    for (int v = 0; v < 8; ++v) d2[v] = sqqr[v] + sqc - 2.0f * acc[v];
    float dmin = d2[0];
#pragma unroll
    for (int v = 1; v < 8; ++v) dmin = fminf(dmin, d2[v]);
    if (__ballot(dmin <= R2) == 0ull) continue;

#pragma unroll
    for (int v = 0; v < 8; ++v) {
      const unsigned m = (unsigned)__ballot(d2[v] <= R2);  // lo16=row v, hi16=row v+8
      if (m == 0u) continue;

      const unsigned mh    = hi ? (m >> 16) : (m & 0xFFFFu);
      const int      r     = hi ? (v + 8) : v;
      const int      bud   = KNN - __shfl(cnt, r);         // remaining budget (>=0)
      const int      rank  = __popc(mh & ((1u << hl) - 1u));
      const int      pop   = __popc(mh);
      const int      taken = pop < bud ? pop : bud;

      float ang = -3.402823466e38f;
      if ((d2[v] <= R2) && (rank < bud)) {                 // accepted: ordered-first-K
        const float gx  = cx - qxr[v];
        const float gy  = cy - qyr[v];
        const float gz  = cz - qzr[v];
        const float crx = nyr[v] * gz - nzr[v] * gy;
        const float cry = nzr[v] * gx - nxr[v] * gz;
        const float crz = nxr[v] * gy - nyr[v] * gx;
        const float cno = sqrtf(crx * crx + cry * cry + crz * crz + PPF_EPS);
        const float dt  = nxr[v] * gx + nyr[v] * gy + nzr[v] * gz;
        ang = atan2f(cno, dt);
      }

      // Max-reduce within each 16-lane half (xor <16 stays in-half).
#pragma unroll
      for (int s = 1; s < 16; s <<= 1)
        ang = fmaxf(ang, __shfl_xor(ang, s));

      const float angO   = __shfl(ang, lane ^ 16);   // other half's row result
      const int   takenO = __shfl(taken, lane ^ 16);
      if (lane == v)            { mx = fmaxf(mx, ang);  cnt += taken;  }
      else if (lane == (v + 8)) { mx = fmaxf(mx, angO); cnt += takenO; }
    }

    // All 16 rows saturated their K budget? (rare: avg ~17 < 32 neighbors)
    // S_ENDPGM's implicit wait-idle covers the one staged async on break.
    if (__ballot(lane < 16 && cnt < KNN) == 0ull) break;
  }

  if (lane < 16) out[(size_t)b * NPTS + tile * 16 + lane] = mx;
}

extern "C" void kernel_launch(void* const* d_in, const int* in_sizes, int n_in,
                              void* d_out, int out_size, void* d_ws, size_t ws_size,
                              hipStream_t stream) {
  (void)in_sizes; (void)n_in; (void)out_size; (void)d_ws; (void)ws_size;
  const float* coords  = (const float*)d_in[0];   // (B,3,N) f32
  const float* normals = (const float*)d_in[1];   // (B,3,N) f32
  float*       out     = (float*)d_out;           // (B,1,N) f32

  dim3 grid(NB * NTILES);   // 4 * 256 = 1024 waves
  dim3 block(32);           // one wave32 per block
  ppf_ball_wmma<<<grid, block, 0, stream>>>(coords, normals, out);
}